// social_stgcn_73272142069882
// MI455X (gfx1250) — compile-verified
//
#include <hip/hip_runtime.h>
#include <hip/hip_bf16.h>
#include <math.h>

// ---------------- problem constants (match reference) ----------------
#define N_NODES 50000
#define N_EDGES 800000
#define FDIM    128        // F_IN == F_OUT
#define HL      24         // LSTM hidden
#define GCOLS   96         // 4 gates * 24
#define ROWT    3125       // 50000 / 16 row tiles (exact)

typedef __attribute__((ext_vector_type(2))) float v2f;
typedef __attribute__((ext_vector_type(8))) float v8f;

// ---------------------------------------------------------------------
// elementwise helpers
// ---------------------------------------------------------------------
__global__ void zero_k(float* p, int n) {
    for (int i = blockIdx.x * blockDim.x + threadIdx.x; i < n;
         i += gridDim.x * blockDim.x)
        p[i] = 0.0f;
}

__global__ void count_deg_k(const int* __restrict__ row,
                            const int* __restrict__ col,
                            float* __restrict__ deg,     // indexed by col
                            float* __restrict__ degc) {  // indexed by row
    int e = blockIdx.x * blockDim.x + threadIdx.x;
    if (e >= N_EDGES) return;
    unsafeAtomicAdd(&deg[col[e]], 1.0f);
    unsafeAtomicAdd(&degc[row[e]], 1.0f);
}

__global__ void node_norms_k(const float* __restrict__ deg,
                             const float* __restrict__ degc,
                             float* __restrict__ dinv,
                             float* __restrict__ selfw,
                             float* __restrict__ dinvc) {
    int i = blockIdx.x * blockDim.x + threadIdx.x;
    if (i >= N_NODES) return;
    float d  = deg[i] + 2.0f;                 // improved=True self-loop weight 2
    float di = rsqrtf(d);
    dinv[i]  = di;
    selfw[i] = 2.0f * di * di;
    float dc = degc[i];
    dinvc[i] = (dc > 0.0f) ? rsqrtf(fmaxf(dc, 1.0f)) : 0.0f;
}

__global__ void edge_norms_k(const int* __restrict__ row,
                             const int* __restrict__ col,
                             const float* __restrict__ dinv,
                             const float* __restrict__ dinvc,
                             float* __restrict__ enorm,
                             float* __restrict__ cnorm) {
    int e = blockIdx.x * blockDim.x + threadIdx.x;
    if (e >= N_EDGES) return;
    int r = row[e], c = col[e];
    enorm[e] = dinv[r] * dinv[c];
    cnorm[e] = -(dinvc[r] * dinvc[c]);
}

// out = hw * selfw[n] + b[f]   (GCN self-loop + bias, overwrites dst)
__global__ void gcn_combine_k(const float* __restrict__ hw,
                              const float* __restrict__ selfw,
                              const float* __restrict__ b,
                              float* __restrict__ out, int total) {
    int i = blockIdx.x * blockDim.x + threadIdx.x;
    if (i >= total) return;
    int n = i >> 7;       // / 128
    int f = i & 127;
    out[i] = hw[i] * selfw[n] + b[f];
}

// dst[col[e]] += src[row[e]] * w[e]    (128 floats/edge; 32 lanes x float4)
__global__ void edge_scatter_k(const float* __restrict__ src,
                               float* __restrict__ dst,
                               const float* __restrict__ w,
                               const int* __restrict__ row,
                               const int* __restrict__ col) {
    int tid = blockIdx.x * blockDim.x + threadIdx.x;
    int e = tid >> 5;
    if (e >= N_EDGES) return;
    int f4 = (tid & 31) << 2;
    int r = row[e], c = col[e];
    float we = w[e];
    const float4 v = *(const float4*)(src + r * FDIM + f4);
    float* d = dst + c * FDIM + f4;
    unsafeAtomicAdd(d + 0, v.x * we);
    unsafeAtomicAdd(d + 1, v.y * we);
    unsafeAtomicAdd(d + 2, v.z * we);
    unsafeAtomicAdd(d + 3, v.w * we);
}

__global__ void relu_k(float* p, int n) {
    int i = blockIdx.x * blockDim.x + threadIdx.x;
    if (i < n) p[i] = fmaxf(p[i], 0.0f);
}

// tx2 = 2*tx2 - tx0  (in place over tmp buffer)
__global__ void cheb2_k(float* __restrict__ tx2, const float* __restrict__ tx0,
                        int n) {
    int i = blockIdx.x * blockDim.x + threadIdx.x;
    if (i < n) tx2[i] = 2.0f * tx2[i] - tx0[i];
}

// repack lstm_Wx (4,3,128,24) -> Bm (384 x 96) with Bm[(k*128+c)*96 + g*24+j]
__global__ void repack_k(const float* __restrict__ Wx, float* __restrict__ Bm) {
    int idx = blockIdx.x * blockDim.x + threadIdx.x;   // source linear index
    if (idx >= 4 * 3 * FDIM * HL) return;
    int j = idx % HL;
    int c = (idx / HL) % FDIM;
    int k = (idx / (HL * FDIM)) % 3;
    int g = idx / (HL * FDIM * 3);
    Bm[(k * FDIM + c) * GCOLS + g * HL + j] = Wx[idx];
}

// ---------------------------------------------------------------------
// WMMA fp32 GEMM:  C[16 x NC row tile] = sum_chunks A_chunk @ B_chunk
// A chunks are N x 128 row-major; B is (nChunks*128) x NC row-major.
// NC is a compile-time constant so all B accesses fold to immediate
// offsets (no per-iteration 64-bit address adds).
// blockDim = 32 * (NC/16); blockIdx.x = row tile (exact tiling, no tails).
// ---------------------------------------------------------------------
template <int NC>
__device__ __forceinline__ v8f gemm_chunk(v8f acc, const float* __restrict__ Ap,
                                          const float* __restrict__ B, int chunk,
                                          int row0, int col0, int m, int kh) {
    const float* arow = Ap + (size_t)(row0 + m) * FDIM + kh;
    const float* bcol = B + (size_t)(chunk * FDIM + kh) * NC + col0 + m;
#pragma unroll
    for (int kc = 0; kc < FDIM; kc += 4) {
        v2f a, b;
        a.x = arow[kc];
        a.y = arow[kc + 1];
        b.x = bcol[kc * NC];            // compile-time immediate offsets
        b.y = bcol[(kc + 1) * NC];
        acc = __builtin_amdgcn_wmma_f32_16x16x4_f32(false, a, false, b,
                                                    (short)0, acc, false, false);
    }
    return acc;
}

template <int NC>
__global__ void wmma_gemm_k(const float* __restrict__ A0,
                            const float* __restrict__ A1,
                            const float* __restrict__ A2,
                            const float* __restrict__ B,
                            float* __restrict__ C) {
    const int lane = threadIdx.x & 31;
    const int wave = threadIdx.x >> 5;
    const int row0 = blockIdx.x << 4;
    const int col0 = wave << 4;
    const int m = lane & 15;            // M (for A) / N (for B) within tile
    const int kh = (lane >> 4) << 1;    // K sub-offset: 0 (lanes 0-15) / 2 (16-31)

    v8f acc = {};
    acc = gemm_chunk<NC>(acc, A0, B, 0, row0, col0, m, kh);
    if (A1) acc = gemm_chunk<NC>(acc, A1, B, 1, row0, col0, m, kh);
    if (A2) acc = gemm_chunk<NC>(acc, A2, B, 2, row0, col0, m, kh);

    // C/D layout: VGPR v -> row = v + 8*(lane>=16), col = lane&15
    const int rbase = row0 + ((lane >> 4) << 3);
    const int cidx = col0 + m;
#pragma unroll
    for (int v = 0; v < 8; ++v)
        C[(size_t)(rbase + v) * NC + cidx] = acc[v];
}

// ---------------------------------------------------------------------
// GConvLSTM pointwise (C0 = H0 = 0 folded) + ReLU + Linear head + round
// pre: N x 96 = [I | F | Tc | O] pre-activations (no biases yet)
// ---------------------------------------------------------------------
__device__ __forceinline__ float sigmoidf_(float x) {
    return 1.0f / (1.0f + expf(-x));
}

__global__ void lstm_head_k(const float* __restrict__ pre,
                            const float* __restrict__ bx,  // (4,24)
                            const float* __restrict__ bh,  // (4,24)
                            const float* __restrict__ bb,  // (4,24)
                            const float* __restrict__ wc,  // (3,24)
                            const float* __restrict__ linW, // (24,3)
                            const float* __restrict__ linb, // (3,)
                            float* __restrict__ out) {
    int n = blockIdx.x * blockDim.x + threadIdx.x;
    if (n >= N_NODES) return;
    const float* p = pre + (size_t)n * GCOLS;
    float a0 = linb[0], a1 = linb[1], a2 = linb[2];
#pragma unroll 4
    for (int j = 0; j < HL; ++j) {
        float bI = bx[j]          + bh[j]          + bb[j];
        float bT = bx[2 * HL + j] + bh[2 * HL + j] + bb[2 * HL + j];
        float bO = bx[3 * HL + j] + bh[3 * HL + j] + bb[3 * HL + j];
        float I = sigmoidf_(p[j] + bI);
        float T = tanhf(p[2 * HL + j] + bT);
        float C = I * T;                               // Fg * C0 == 0
        float O = sigmoidf_(p[3 * HL + j] + wc[2 * HL + j] * C + bO);
        float H = O * tanhf(C);
        float r = fmaxf(H, 0.0f);
        a0 += r * linW[j * 3 + 0];
        a1 += r * linW[j * 3 + 1];
        a2 += r * linW[j * 3 + 2];
    }
    out[n * 3 + 0] = rintf(a0);   // round half-to-even == jnp.round
    out[n * 3 + 1] = rintf(a1);
    out[n * 3 + 2] = rintf(a2);
}

// ---------------------------------------------------------------------
// launch
// ---------------------------------------------------------------------
extern "C" void kernel_launch(void* const* d_in, const int* in_sizes, int n_in,
                              void* d_out, int out_size, void* d_ws,
                              size_t ws_size, hipStream_t stream) {
    const float* x    = (const float*)d_in[0];
    const int*   ei   = (const int*)d_in[1];
    const int*   row  = ei;
    const int*   col  = ei + N_EDGES;
    const float* gcnW = (const float*)d_in[2];   // (3,128,128)
    const float* gcnB = (const float*)d_in[3];   // (3,128)
    const float* Wx   = (const float*)d_in[4];   // (4,3,128,24)
    const float* bx   = (const float*)d_in[5];   // (4,24)
    // d_in[6] = lstm_Wh: dead (H0 == 0 -> cheb(H0,Wh,bh) == bh)
    const float* bh   = (const float*)d_in[7];   // (4,24)
    const float* bb   = (const float*)d_in[8];   // (4,24)
    const float* wc   = (const float*)d_in[9];   // (3,24)
    const float* linW = (const float*)d_in[10];  // (24,3)
    const float* linb = (const float*)d_in[11];  // (3,)
    float* out = (float*)d_out;

    // workspace carve-up (all chunk sizes multiples of 4 floats -> 16B aligned)
    float* ws    = (float*)d_ws;
    float* deg   = ws;                 ws += N_NODES;
    float* degc  = ws;                 ws += N_NODES;
    float* dinv  = ws;                 ws += N_NODES;
    float* selfw = ws;                 ws += N_NODES;
    float* dinvc = ws;                 ws += N_NODES;
    float* enorm = ws;                 ws += N_EDGES;
    float* cnorm = ws;                 ws += N_EDGES;
    float* bufA  = ws;                 ws += (size_t)N_NODES * FDIM;
    float* bufB  = ws;                 ws += (size_t)N_NODES * FDIM;
    float* bufHW = ws;                 ws += (size_t)N_NODES * FDIM;
    float* pre   = ws;                 ws += (size_t)N_NODES * GCOLS;
    float* Bm    = ws;                 ws += 3 * FDIM * GCOLS;
    size_t need = (size_t)(ws - (float*)d_ws) * sizeof(float);
    if (ws_size < need) return;

    const int NF = N_NODES * FDIM;
    dim3 b256(256);
    dim3 gE((N_EDGES + 255) / 256);
    dim3 gN((N_NODES + 255) / 256);
    dim3 gNF((NF + 255) / 256);
    dim3 gSc(((size_t)N_EDGES * 32 + 255) / 256);

    // ---- degrees + norms ----
    zero_k<<<dim3((2 * N_NODES + 255) / 256), b256, 0, stream>>>(deg, 2 * N_NODES);
    count_deg_k<<<gE, b256, 0, stream>>>(row, col, deg, degc);
    node_norms_k<<<gN, b256, 0, stream>>>(deg, degc, dinv, selfw, dinvc);
    edge_norms_k<<<gE, b256, 0, stream>>>(row, col, dinv, dinvc, enorm, cnorm);

    // ---- 3 stacked GCN layers (ping-pong x -> A -> B -> A) ----
    const float* hin[3] = {x, bufA, bufB};
    float*       hout[3] = {bufA, bufB, bufA};
    for (int l = 0; l < 3; ++l) {
        wmma_gemm_k<FDIM><<<dim3(ROWT), dim3(256), 0, stream>>>(
            hin[l], nullptr, nullptr, gcnW + (size_t)l * FDIM * FDIM, bufHW);
        gcn_combine_k<<<gNF, b256, 0, stream>>>(bufHW, selfw,
                                                gcnB + (size_t)l * FDIM, hout[l], NF);
        edge_scatter_k<<<gSc, b256, 0, stream>>>(bufHW, hout[l], enorm, row, col);
    }
    relu_k<<<gNF, b256, 0, stream>>>(bufA, NF);            // Tx0 = relu(h3)

    // ---- Chebyshev basis: Tx1 = S Tx0 ; Tx2 = 2 S Tx1 - Tx0 ----
    zero_k<<<gNF, b256, 0, stream>>>(bufHW, NF);
    edge_scatter_k<<<gSc, b256, 0, stream>>>(bufA, bufHW, cnorm, row, col);  // Tx1
    zero_k<<<gNF, b256, 0, stream>>>(bufB, NF);
    edge_scatter_k<<<gSc, b256, 0, stream>>>(bufHW, bufB, cnorm, row, col);  // S Tx1
    cheb2_k<<<gNF, b256, 0, stream>>>(bufB, bufA, NF);                       // Tx2

    // ---- fused 4-gate x 3-order GEMM: pre = [Tx0|Tx1|Tx2] @ Bm (384x96) ----
    repack_k<<<dim3((4 * 3 * FDIM * HL + 255) / 256), b256, 0, stream>>>(Wx, Bm);
    wmma_gemm_k<GCOLS><<<dim3(ROWT), dim3(192), 0, stream>>>(bufA, bufHW, bufB,
                                                             Bm, pre);

    // ---- LSTM pointwise + head ----
    lstm_head_k<<<gN, b256, 0, stream>>>(pre, bx, bh, bb, wc, linW, linb, out);
}